// PointSMCAPriorBranch_80771154968919
// MI455X (gfx1250) — compile-verified
//
#include <hip/hip_runtime.h>

#define BATCH 8
#define H 320
#define W 320
#define HW (H*W)
#define TOPK 80

typedef float v2f __attribute__((ext_vector_type(2)));
typedef float v8f __attribute__((ext_vector_type(8)));

// Order-preserving float->uint mapping so uint atomicMin/Max == float min/max.
__device__ __forceinline__ unsigned mapf(float f) {
  unsigned u = __float_as_uint(f);
  return (u & 0x80000000u) ? ~u : (u | 0x80000000u);
}
__device__ __forceinline__ float unmapf(unsigned u) {
  return (u & 0x80000000u) ? __uint_as_float(u & 0x7FFFFFFFu) : __uint_as_float(~u);
}

__global__ void k_init(unsigned* red) {
  int i = threadIdx.x;
  if (i < BATCH) {
    red[i]           = 0xFFFFFFFFu; // Pmin
    red[i + BATCH]   = 0u;          // Pmax
    red[i + 2*BATCH] = 0xFFFFFFFFu; // Gmin
    red[i + 3*BATCH] = 0u;          // Gmax
  }
}

// 9x9 avg pools (zero pad, /81) of occ and score, P = local_occ*local_mass,
// plus per-batch min/max of P. Separable sums in LDS, 32x32 output tiles.
__global__ void k_pool(const float* __restrict__ score, float* __restrict__ P,
                       unsigned* redPmin, unsigned* redPmax) {
  const int bb = blockIdx.z;
  const int by = blockIdx.y * 32;
  const int bx = blockIdx.x * 32;
  __shared__ float tile[40][40];
  __shared__ float hs[40][32];
  __shared__ float ho[40][32];
  __shared__ unsigned rmn[256], rmx[256];
  const int tid = threadIdx.x;
  const float* sb = score + bb * HW;
  for (int i = tid; i < 1600; i += 256) {
    int r = i / 40, c = i % 40;
    int gy = by + r - 4, gx = bx + c - 4;
    float v = 0.0f;
    if (gy >= 0 && gy < H && gx >= 0 && gx < W) v = sb[gy * W + gx];
    tile[r][c] = v;
  }
  __syncthreads();
  for (int i = tid; i < 1280; i += 256) {
    int r = i / 32, c = i % 32;
    float ss = 0.0f, so = 0.0f;
#pragma unroll
    for (int t = 0; t < 9; ++t) {
      float v = tile[r][c + t];
      ss += v;
      so += (v > 0.2f) ? 1.0f : 0.0f;
    }
    hs[r][c] = ss; ho[r][c] = so;
  }
  __syncthreads();
  float lmin = 3.4e38f, lmax = -3.4e38f;
  for (int i = tid; i < 1024; i += 256) {
    int ly = i / 32, lx = i % 32;
    float vs = 0.0f, vo = 0.0f;
#pragma unroll
    for (int j = 0; j < 9; ++j) { vs += hs[ly + j][lx]; vo += ho[ly + j][lx]; }
    float p = (vo * (1.0f / 81.0f)) * (vs * (1.0f / 81.0f));
    P[bb * HW + (by + ly) * W + (bx + lx)] = p;
    lmin = fminf(lmin, p); lmax = fmaxf(lmax, p);
  }
  rmn[tid] = mapf(lmin); rmx[tid] = mapf(lmax);
  __syncthreads();
  for (int s = 128; s > 0; s >>= 1) {
    if (tid < s) {
      rmn[tid] = min(rmn[tid], rmn[tid + s]);
      rmx[tid] = max(rmx[tid], rmx[tid + s]);
    }
    __syncthreads();
  }
  if (tid == 0) { atomicMin(&redPmin[bb], rmn[0]); atomicMax(&redPmax[bb], rmx[0]); }
}

// compact_score = score * norm01(P); 5x5 NMS (-inf pad): maxima = cs*(cs==maxpool5)
__global__ void k_nms(const float* __restrict__ score, const float* __restrict__ P,
                      const unsigned* redPmin, const unsigned* redPmax,
                      float* __restrict__ maxima) {
  const int bb = blockIdx.z;
  const int by = blockIdx.y * 32, bx = blockIdx.x * 32;
  __shared__ float cs[36][36];
  const int tid = threadIdx.x;
  float mn = unmapf(redPmin[bb]);
  float mx = unmapf(redPmax[bb]);
  float inv = 1.0f / (mx - mn + 1e-6f);
  const float* sb = score + bb * HW;
  const float* pb = P + bb * HW;
  for (int i = tid; i < 1296; i += 256) {
    int r = i / 36, c = i % 36;
    int gy = by + r - 2, gx = bx + c - 2;
    float v = -3.4e38f;
    if (gy >= 0 && gy < H && gx >= 0 && gx < W) {
      int o = gy * W + gx;
      v = sb[o] * ((pb[o] - mn) * inv);
    }
    cs[r][c] = v;
  }
  __syncthreads();
  for (int i = tid; i < 1024; i += 256) {
    int ly = i / 32, lx = i % 32;
    float c0 = cs[ly + 2][lx + 2];
    float m = -3.4e38f;
#pragma unroll
    for (int dy = 0; dy < 5; ++dy)
#pragma unroll
      for (int dx = 0; dx < 5; ++dx)
        m = fmaxf(m, cs[ly + dy][lx + dx]);
    maxima[bb * HW + (by + ly) * W + (bx + lx)] = (c0 == m) ? c0 : 0.0f;
  }
}

// Top-80 per batch: 80 iterated argmax reductions with destructive masking.
// Emits per-peak exp2-folded params:
//   c1 = log2(e)/(2*sigma^2+1e-6), c0 = log2(val)   (val==0 -> -inf -> exp2 -> 0)
// peaks[k] = (c0, xs, ys, c1)
__global__ void __launch_bounds__(1024) k_topk(float* __restrict__ maxima,
                                               const float* __restrict__ depth,
                                               float4* __restrict__ peaks) {
  const int bb = blockIdx.x;
  const int tid = threadIdx.x;
  __shared__ float sv[1024];
  __shared__ int si[1024];
  float* mb = maxima + bb * HW;
  const float* db = depth + bb * HW;
  for (int k = 0; k < TOPK; ++k) {
    float bv = -3.4e38f; int bi = 0;
    for (int i = tid; i < HW; i += 1024) {
      __builtin_prefetch(mb + i + 4096, 0, 0);   // gfx1250 global_prefetch_b8
      float v = mb[i];
      if (v > bv || (v == bv && i < bi)) { bv = v; bi = i; }
    }
    sv[tid] = bv; si[tid] = bi;
    __syncthreads();
    for (int s = 512; s > 0; s >>= 1) {
      if (tid < s) {
        float v2 = sv[tid + s]; int i2 = si[tid + s];
        if (v2 > sv[tid] || (v2 == sv[tid] && i2 < si[tid])) { sv[tid] = v2; si[tid] = i2; }
      }
      __syncthreads();
    }
    if (tid == 0) {
      float v = sv[0]; int idx = si[0];
      float val = v > 0.0f ? v : 0.0f;
      float xs = (float)(idx % W), ys = (float)(idx / W);
      float z = fmaxf(db[idx], 1e-3f);
      float rad = fminf(fmaxf(14.0f / z, 1.5f), 18.0f);
      float s2 = 0.6f * rad; s2 = s2 * s2;
      float invs = 1.0f / (2.0f * s2 + 1e-6f);
      float c1 = invs * 1.44269504088896340736f;  // inv * log2(e)
      float c0 = __log2f(val);                     // log2(0) = -inf
      peaks[bb * TOPK + k] = make_float4(c0, xs, ys, c1);
      mb[idx] = -1.0f; // remove
    }
    __threadfence_block();
    __syncthreads();
  }
}

// Gaussian splat via V_WMMA_F32_16X16X4_F32, peaks-as-A / pixels-as-B so the
// peak-max reduction is in-lane:
//   D[m][n] = (2*xs_m*c1_m)*x_n + (2*ys_m*c1_m)*y + (-c1_m)*(x_n^2+y^2)
//             + (c0_m - c1_m*(xs_m^2+ys_m^2))
//           = c0_m - c1_m*dist^2(m,n)   ;  contribution = exp2(D)
// A(16x4): a[0] = lane<16 ? A[m][0] : A[m][2]; a[1] = A[m][1] | A[m][3]; m=lane%16
// B(4x16): b[0] = lane<16 ? B[0][n]=x_n : B[2][n]=x_n^2+y^2; b[1] = y | 1; n=lane%16
// C = 0 (inline). D: lane holds pixel n=lane%16, peaks m=r+8*(lane/16) over 8 VGPRs.
// max over group = in-lane max over r, halves merged with a single shfl_xor(16).
__global__ void __launch_bounds__(256) k_splat(const float4* __restrict__ peaks,
                                               float* __restrict__ gauss,
                                               unsigned* redGmin, unsigned* redGmax) {
  const int blk = blockIdx.x;            // B*H/8 blocks; 8 waves = 8 rows each
  const int bb = blk / (H / 8);
  const int y0 = (blk % (H / 8)) * 8;
  const int tid = threadIdx.x;
  const int lane = tid & 31;
  const int wave = tid >> 5;
  const int y = y0 + wave;
  const int nlo = lane & 15;
  const bool hi = lane >= 16;

  v2f av[5];
#pragma unroll
  for (int g = 0; g < 5; ++g) {
    float4 p = peaks[bb * TOPK + g * 16 + nlo];
    float c0 = p.x, xs = p.y, ys = p.z, c1 = p.w;
    av[g][0] = hi ? (-c1) : (2.0f * xs * c1);
    av[g][1] = hi ? fmaf(-c1, xs * xs + ys * ys, c0) : (2.0f * ys * c1);
  }
  float fy = (float)y;
  float y2 = fy * fy;
  v2f b;
  b[1] = hi ? 1.0f : fy;
  v8f cz = {0.0f, 0.0f, 0.0f, 0.0f, 0.0f, 0.0f, 0.0f, 0.0f};
  float wmin = 3.4e38f, wmax = -3.4e38f;
  float* grow = gauss + bb * HW + y * W;

  for (int seg = 0; seg < W / 16; ++seg) {
    int xbase = seg * 16;
    float xn = (float)(xbase + nlo);
    b[0] = hi ? fmaf(xn, xn, y2) : xn;
    float vmax = 0.0f;
#pragma unroll
    for (int g = 0; g < 5; ++g) {
      v8f d = __builtin_amdgcn_wmma_f32_16x16x4_f32(
          false, av[g], false, b, (short)0, cz, false, false);
#pragma unroll
      for (int r = 0; r < 8; ++r)
        vmax = fmaxf(vmax, __builtin_amdgcn_exp2f(d[r]));
    }
    // merge the two 16-lane halves (peaks m=0..7 vs m=8..15 of each group)
    vmax = fmaxf(vmax, __shfl_xor(vmax, 16, 32));
    if (!hi) grow[xbase + nlo] = vmax;   // 16 consecutive floats, coalesced
    wmin = fminf(wmin, vmax); wmax = fmaxf(wmax, vmax);
  }

  __shared__ unsigned rmn[256], rmx[256];
  rmn[tid] = mapf(wmin); rmx[tid] = mapf(wmax);
  __syncthreads();
  for (int s = 128; s > 0; s >>= 1) {
    if (tid < s) {
      rmn[tid] = min(rmn[tid], rmn[tid + s]);
      rmx[tid] = max(rmx[tid], rmx[tid + s]);
    }
    __syncthreads();
  }
  if (tid == 0) { atomicMin(&redGmin[bb], rmn[0]); atomicMax(&redGmax[bb], rmx[0]); }
}

__global__ void k_norm(float* __restrict__ out, const unsigned* redGmin,
                       const unsigned* redGmax) {
  int i = blockIdx.x * blockDim.x + threadIdx.x;
  if (i >= BATCH * HW) return;
  int bb = i / HW;
  float mn = unmapf(redGmin[bb]);
  float mx = unmapf(redGmax[bb]);
  float inv = 1.0f / (mx - mn + 1e-6f);
  out[i] = (out[i] - mn) * inv;
}

extern "C" void kernel_launch(void* const* d_in, const int* in_sizes, int n_in,
                              void* d_out, int out_size, void* d_ws, size_t ws_size,
                              hipStream_t stream) {
  (void)in_sizes; (void)n_in; (void)out_size; (void)ws_size;
  const float* score = (const float*)d_in[0];
  const float* depth = (const float*)d_in[1];
  float* out = (float*)d_out;

  float* P       = (float*)d_ws;                      // B*HW
  float* maxima  = P + BATCH * HW;                    // B*HW
  float4* peaks  = (float4*)(maxima + BATCH * HW);    // B*TOPK (offset 16B aligned)
  unsigned* red  = (unsigned*)(peaks + BATCH * TOPK); // 4*B uints

  k_init<<<dim3(1), dim3(32), 0, stream>>>(red);
  dim3 gtile(W / 32, H / 32, BATCH);
  k_pool<<<gtile, 256, 0, stream>>>(score, P, red, red + BATCH);
  k_nms<<<gtile, 256, 0, stream>>>(score, P, red, red + BATCH, maxima);
  k_topk<<<dim3(BATCH), 1024, 0, stream>>>(maxima, depth, peaks);
  k_splat<<<dim3(BATCH * H / 8), 256, 0, stream>>>(peaks, out, red + 2 * BATCH, red + 3 * BATCH);
  int n = BATCH * HW;
  k_norm<<<dim3((n + 255) / 256), 256, 0, stream>>>(out, red + 2 * BATCH, red + 3 * BATCH);
}